// GNNDiffMaskGate_69071664054322
// MI455X (gfx1250) — compile-verified
//
#include <hip/hip_runtime.h>
#include <hip/hip_bf16.h>
#include <math.h>

typedef __attribute__((ext_vector_type(2))) float v2f;
typedef __attribute__((ext_vector_type(8))) float v8f;

#define H      256
#define HA     64
#define SEQ    512
#define BZ     4
#define MROWS  (BZ*SEQ)      // 2048 rows in all GEMMs
#define CATLD  512           // leading dim of dense-cat activation buffer
#define TEMP_  0.2f
#define LO_    (-0.2f)
#define RO_    1.0f
#define MAXACT 10.0f
// c0 = TEMP*log(-LO/RO) = 0.2*ln(0.2)
#define C0_    (-0.32188758248682006f)

// ---------------------------------------------------------------------------
// Generic fp32 WMMA GEMM: C[2048 x N] = op(A [+A2]) * B with fused epilogues.
// One wave32 computes a 16(M) x 64(N) strip: A-frag loaded once per K-step and
// reused by 4 independent V_WMMA_F32_16X16X4_F32 (co-executing XDL issues,
// EXEC all-1s). Per-column weight-norm scaling is applied in the EPILOGUE
// (column scaling commutes with the GEMM), keeping the K-loop pure load+WMMA.
// A-frag (16x4 f32): lanes 0-15 -> M=lane, VGPR0=K0,VGPR1=K1;
//                    lanes 16-31 -> M=lane-16, VGPR0=K2,VGPR1=K3.
// B-frag (4x16) mirrored; C/D: VGPR i -> M = i + 8*(lane>=16), N = lane&15.
// ---------------------------------------------------------------------------
struct GemmDesc {
  const float* A;  int lda;  int aoff;      // A[m][k] = A[m*lda + aoff + k]
  const float* A2; int lda2; int a2off;     // optional elementwise add on A
  const float* B;  int ldb;  int boff;
  long long bbatch;                         // elem stride per batch b=m>>9 (non-trans B)
  const float* bscale;                      // per-n output scale (weight-norm)
  const float* bias; float bias_mult;       // per-n bias
  const float* denom;                       // per-m divisor
  int relu;
  float* C; int ldc; int coff;
  int store_acolT;                          // store C transposed: [b][n][r]
  int N; int K;
};

template<bool HAS_A2, bool BTRANS>
__global__ __launch_bounds__(128)
void gemm_wmma_kernel(GemmDesc d) {
  const int lane = threadIdx.x & 31;
  const int lidx = lane & 15;
  const int koff = (lane >> 4) * 2;
  const int wgid = blockIdx.x * 4 + (threadIdx.x >> 5);
  const int n64  = d.N >> 6;                // N is always a multiple of 64
  const int m0   = (wgid / n64) << 4;
  const int n0   = (wgid % n64) << 6;
  const int b    = m0 >> 9;                 // batch index (512 rows/batch)

  const float* Arow  = d.A + (size_t)(m0 + lidx) * d.lda + d.aoff;
  const float* A2row = HAS_A2 ? (d.A2 + (size_t)(m0 + lidx) * d.lda2 + d.a2off)
                              : nullptr;
  const float* Bp[4];
  const float* Bbase = nullptr;
#pragma unroll
  for (int nn = 0; nn < 4; ++nn) {
    if (BTRANS) Bp[nn] = d.B + (size_t)(n0 + nn * 16 + lidx) * d.ldb + d.boff;
  }
  if (!BTRANS)
    Bbase = d.B + (size_t)b * d.bbatch + d.boff + n0 + lidx;

  v8f acc[4] = {};
  for (int k0 = 0; k0 < d.K; k0 += 4) {
    const int ka = k0 + koff;
    __builtin_prefetch(Arow + ka + 64, 0, 3);   // global_prefetch_b8, WGP scope
    v2f av;
    av.x = Arow[ka];
    av.y = Arow[ka + 1];
    if (HAS_A2) { av.x += A2row[ka]; av.y += A2row[ka + 1]; }
    v2f bv[4];
    if (BTRANS) {
#pragma unroll
      for (int nn = 0; nn < 4; ++nn) {
        bv[nn].x = Bp[nn][ka];
        bv[nn].y = Bp[nn][ka + 1];
      }
    } else {
      const float* q = Bbase + (size_t)ka * d.ldb;
#pragma unroll
      for (int nn = 0; nn < 4; ++nn) {
        bv[nn].x = q[nn * 16];
        bv[nn].y = q[nn * 16 + d.ldb];
      }
    }
#pragma unroll
    for (int nn = 0; nn < 4; ++nn)
      acc[nn] = __builtin_amdgcn_wmma_f32_16x16x4_f32(false, av, false, bv[nn],
                                                      (short)0, acc[nn],
                                                      false, false);
  }

  // epilogue (runs once; runtime flags fine here)
  const int rbase = m0 + ((lane >> 4) << 3);
  float dinv[8];
  if (d.denom) {
#pragma unroll
    for (int i = 0; i < 8; ++i) dinv[i] = 1.f / d.denom[rbase + i];
  }
#pragma unroll
  for (int nn = 0; nn < 4; ++nn) {
    const int col = n0 + nn * 16 + lidx;
    const float cscale = d.bscale ? d.bscale[col] : 1.f;
    const float biasv  = d.bias ? d.bias_mult * d.bias[col] : 0.f;
#pragma unroll
    for (int i = 0; i < 8; ++i) {
      const int row = rbase + i;
      float v = acc[nn][i] * cscale + biasv;
      if (d.denom) v *= dinv[i];
      if (d.relu)  v = fmaxf(v, 0.f);
      if (d.store_acolT) {
        const int bb = row >> 9, r = row & 511;
        d.C[((size_t)bb * HA + col) * SEQ + r] = v;
      } else {
        d.C[(size_t)row * d.ldc + d.coff + col] = v;
      }
    }
  }
}

// ---------------------------------------------------------------------------
// Weight-norm precompute
// ---------------------------------------------------------------------------
__global__ __launch_bounds__(64)
void wn1_kernel(const float* v, const float* g, float* scale) {
  const int a = threadIdx.x;                 // 64 rows, 512 cols each
  float s = 0.f;
  for (int h = 0; h < 2 * H; ++h) { const float x = v[a * 2 * H + h]; s += x * x; }
  scale[a] = g[a] / sqrtf(s);
}

__global__ __launch_bounds__(64)
void wn2_kernel(const float* v, const float* g, float* w2n) {
  __shared__ float red[64];
  const int a = threadIdx.x;
  const float x = v[a];
  red[a] = x * x; __syncthreads();
  for (int o = 32; o > 0; o >>= 1) { if (a < o) red[a] += red[a + o]; __syncthreads(); }
  w2n[a] = v[a] * (g[0] / sqrtf(red[0]));
}

// ---------------------------------------------------------------------------
// Fused pairwise gate: logits[b,r,c] = tanh(sum_a W2n[a]*tanh(ar+ac+b1)) * 10 + bias
// then HardConcrete rsample -> gates. One block per (b,r) row.
// ---------------------------------------------------------------------------
__global__ __launch_bounds__(256)
void logits_gates_kernel(const float* __restrict__ arow,
                         const float* __restrict__ acolT,
                         const float* __restrict__ w2n,
                         const float* __restrict__ g1b,
                         const float* __restrict__ g2b_p,
                         const float* __restrict__ bias_p,
                         const float* __restrict__ noise,
                         float* __restrict__ logits,
                         float* __restrict__ gates) {
  __shared__ float sa[HA], sw[HA], sb[HA];
  const int m = blockIdx.x;                  // 0..2047 = (b, r)
  const int b = m >> 9;
  const int t = threadIdx.x;
  if (t < HA) { sa[t] = arow[(size_t)m * HA + t]; sw[t] = w2n[t]; sb[t] = g1b[t]; }
  __syncthreads();
  const float g2b = g2b_p[0], gb = bias_p[0];
  const float* ac = acolT + (size_t)b * HA * SEQ;
  for (int c = t; c < SEQ; c += 256) {
    float s = 0.f;
#pragma unroll 8
    for (int a = 0; a < HA; ++a)
      s += sw[a] * tanhf(sa[a] + ac[a * SEQ + c] + sb[a]);
    const float logit = tanhf(s + g2b) * MAXACT + gb;
    const size_t idx = (size_t)m * SEQ + c;
    logits[idx] = logit;
    const float u = noise[idx];
    const float z = (logf(u) - log1pf(-u) + logit) * (1.f / TEMP_);
    const float sig = 1.f / (1.f + expf(-z));
    const float gv = sig * (RO_ - LO_) + LO_;
    gates[idx] = fminf(fmaxf(gv, 0.f), 1.f);
  }
}

__global__ __launch_bounds__(256)
void denom_kernel(const float* __restrict__ gates, float* __restrict__ denom) {
  __shared__ float red[256];
  const int m = blockIdx.x, t = threadIdx.x;
  red[t] = gates[(size_t)m * SEQ + t] + gates[(size_t)m * SEQ + t + 256];
  __syncthreads();
  for (int o = 128; o > 0; o >>= 1) { if (t < o) red[t] += red[t + o]; __syncthreads(); }
  if (t == 0) denom[m] = red[0] + 1.f;
}

// ---------------------------------------------------------------------------
// L0 loss: rowsum[m] = sum_c -softplus(c0 - logits), then masked per-batch mean
// ---------------------------------------------------------------------------
__global__ __launch_bounds__(256)
void l0_rowsum_kernel(const float* __restrict__ logits, float* __restrict__ rowsum) {
  __shared__ float red[256];
  const int m = blockIdx.x, t = threadIdx.x;
  float s = 0.f;
  for (int c = t; c < SEQ; c += 256) {
    const float x = C0_ - logits[(size_t)m * SEQ + c];
    const float sp = (x > 20.f) ? x : log1pf(expf(x));
    s -= sp;
  }
  red[t] = s; __syncthreads();
  for (int o = 128; o > 0; o >>= 1) { if (t < o) red[t] += red[t + o]; __syncthreads(); }
  if (t == 0) rowsum[m] = red[0];
}

__global__ __launch_bounds__(256)
void l0_final_kernel(const float* __restrict__ rowsum,
                     const float* __restrict__ tm, float* __restrict__ out) {
  __shared__ float rn[256], rd[256];
  const int t = threadIdx.x;
  float total = 0.f;
  for (int b = 0; b < BZ; ++b) {
    float n = 0.f, dsum = 0.f;
    for (int r = t; r < SEQ; r += 256) {
      const float w = tm[b * SEQ + r];
      n += w * rowsum[b * SEQ + r];
      dsum += w;
    }
    rn[t] = n; rd[t] = dsum; __syncthreads();
    for (int o = 128; o > 0; o >>= 1) {
      if (t < o) { rn[t] += rn[t + o]; rd[t] += rd[t + o]; }
      __syncthreads();
    }
    if (t == 0) total += rn[0] / (rd[0] * (float)SEQ);
    __syncthreads();
  }
  if (t == 0) out[0] = total;
}

__global__ __launch_bounds__(256)
void catinit_kernel(const float* __restrict__ hid, float* __restrict__ cat) {
  const int idx = blockIdx.x * 256 + threadIdx.x;   // 2048*256 elements
  const int m = idx >> 8, h = idx & 255;
  cat[(size_t)m * CATLD + h] = hid[idx];
}

// ---------------------------------------------------------------------------
template<bool HAS_A2, bool BTRANS>
static inline void launch_gemm(const GemmDesc& d, hipStream_t s) {
  const int tiles64 = (MROWS / 16) * (d.N / 64);    // 128 * N/64, /4 always exact
  gemm_wmma_kernel<HAS_A2, BTRANS>
      <<<dim3(tiles64 / 4), dim3(128), 0, s>>>(d);
}

extern "C" void kernel_launch(void* const* d_in, const int* in_sizes, int n_in,
                              void* d_out, int out_size, void* d_ws, size_t ws_size,
                              hipStream_t stream) {
  (void)in_sizes; (void)n_in; (void)out_size; (void)ws_size;

  const float* hid   = (const float*)d_in[0];
  const float* tmask = (const float*)d_in[1];
  const float* noise = (const float*)d_in[2];
  const float* g1_v  = (const float*)d_in[3];
  const float* g1_g  = (const float*)d_in[4];
  const float* g1_b  = (const float*)d_in[5];
  const float* g2_v  = (const float*)d_in[6];
  const float* g2_g  = (const float*)d_in[7];
  const float* g2_b  = (const float*)d_in[8];
  const float* gbias = (const float*)d_in[9];
  const float* agg_w = (const float*)d_in[42];
  const float* agg_b = (const float*)d_in[43];

  static const int WIDX[4][4] = {{10,12,-1,-1},{16,18,20,22},{26,28,-1,-1},{32,34,36,38}};
  static const int OIDX[4]    = {14, 24, 30, 40};
  static const int SUBL[4]    = {2, 4, 2, 4};
  static const int HD[4]      = {128, 64, 128, 64};

  // workspace carve-up (floats)
  float* ws      = (float*)d_ws;
  float* arow    = ws;                    size_t off = (size_t)MROWS * HA;        // 131072
  float* acolT   = ws + off;              off += (size_t)MROWS * HA;              // 131072
  float* scale1  = ws + off;              off += 64;
  float* w2n     = ws + off;              off += 64;
  float* logits  = ws + off;              off += (size_t)MROWS * SEQ;             // 1M
  float* gates   = ws + off;              off += (size_t)MROWS * SEQ;             // 1M
  float* denomv  = ws + off;              off += MROWS;
  float* rowsum  = ws + off;              off += MROWS;
  float* cat     = ws + off;              off += (size_t)MROWS * CATLD;           // 1M
  float* Ax      = ws + off;              off += (size_t)MROWS * CATLD;           // 1M
  float* layers  = ws + off;              off += (size_t)MROWS * 4 * H;           // 2M

  // weight-norm scales + dense-cat base (hidden_states) once
  wn1_kernel<<<dim3(1), dim3(64), 0, stream>>>(g1_v, g1_g, scale1);
  wn2_kernel<<<dim3(1), dim3(64), 0, stream>>>(g2_v, g2_g, w2n);
  catinit_kernel<<<dim3(MROWS), dim3(256), 0, stream>>>(hid, cat);

  for (int i = 0; i < 4; ++i) {
    // x for gate logits: hidden_states on layer 0, previous layer output after
    const float* xA = (i == 0) ? hid : layers;
    const int xlda  = (i == 0) ? H : 4 * H;
    const int xoff  = (i == 0) ? 0 : (i - 1) * H;

    // arow = x @ W1[:, :H]^T (weight-norm scale applied in epilogue)
    {
      GemmDesc d = {};
      d.A = xA; d.lda = xlda; d.aoff = xoff;
      d.B = g1_v; d.ldb = 2 * H; d.boff = 0;
      d.bscale = scale1;
      d.C = arow; d.ldc = HA; d.coff = 0;
      d.N = HA; d.K = H;
      launch_gemm<false, true>(d, stream);
    }
    // acolT[b][a][s] = (x @ W1[:, H:]^T) transposed store
    {
      GemmDesc d = {};
      d.A = xA; d.lda = xlda; d.aoff = xoff;
      d.B = g1_v; d.ldb = 2 * H; d.boff = H;
      d.bscale = scale1;
      d.C = acolT; d.store_acolT = 1;
      d.N = HA; d.K = H;
      launch_gemm<false, true>(d, stream);
    }

    logits_gates_kernel<<<dim3(MROWS), dim3(256), 0, stream>>>(
        arow, acolT, w2n, g1_b, g2_b, gbias,
        noise + (size_t)i * BZ * SEQ * SEQ, logits, gates);
    denom_kernel<<<dim3(MROWS), dim3(256), 0, stream>>>(gates, denomv);

    const int subl = SUBL[i], hd = HD[i];
    for (int j = 0; j < subl; ++j) {
      const int F = H + hd * j;
      const float* wj = (const float*)d_in[WIDX[i][j]];
      const float* bj = (const float*)d_in[WIDX[i][j] + 1];
      // Ax = gates @ cat[:, :F]   (batched, non-transposed B)
      {
        GemmDesc d = {};
        d.A = gates; d.lda = SEQ; d.aoff = 0;
        d.B = cat; d.ldb = CATLD; d.boff = 0;
        d.bbatch = (long long)SEQ * CATLD;
        d.C = Ax; d.ldc = CATLD; d.coff = 0;
        d.N = F; d.K = SEQ;
        launch_gemm<false, false>(d, stream);
      }
      // g = relu(((Ax + cat) @ w^T + 2b) / denom) -> cat[:, H + hd*j : ...]
      {
        GemmDesc d = {};
        d.A = Ax; d.lda = CATLD; d.aoff = 0;
        d.A2 = cat; d.lda2 = CATLD; d.a2off = 0;
        d.B = wj; d.ldb = F; d.boff = 0;
        d.bias = bj; d.bias_mult = 2.f;
        d.denom = denomv; d.relu = 1;
        d.C = cat; d.ldc = CATLD; d.coff = H + hd * j;
        d.N = hd; d.K = F;
        launch_gemm<true, true>(d, stream);
      }
    }

    // layer_out = (cat[:, H:2H] + hidden) @ out_w^T + out_b -> layers[:, i*H:]
    {
      const float* ow = (const float*)d_in[OIDX[i]];
      const float* ob = (const float*)d_in[OIDX[i] + 1];
      GemmDesc d = {};
      d.A = cat; d.lda = CATLD; d.aoff = H;
      d.A2 = hid; d.lda2 = H; d.a2off = 0;
      d.B = ow; d.ldb = H; d.boff = 0;
      d.bias = ob; d.bias_mult = 1.f;
      d.C = layers; d.ldc = 4 * H; d.coff = i * H;
      d.N = H; d.K = H;
      launch_gemm<true, true>(d, stream);
    }
  }

  // L0 scalar from last layer's logits
  l0_rowsum_kernel<<<dim3(MROWS), dim3(256), 0, stream>>>(logits, rowsum);
  l0_final_kernel<<<dim3(1), dim3(256), 0, stream>>>(
      rowsum, tmask, (float*)d_out + (size_t)MROWS * H);

  // output = layers @ agg_w^T + agg_b
  {
    GemmDesc d = {};
    d.A = layers; d.lda = 4 * H; d.aoff = 0;
    d.B = agg_w; d.ldb = 4 * H; d.boff = 0;
    d.bias = agg_b; d.bias_mult = 1.f;
    d.C = (float*)d_out; d.ldc = H; d.coff = 0;
    d.N = H; d.K = 4 * H;
    launch_gemm<false, true>(d, stream);
  }
}